// TripletCenterLoss_15917148799612
// MI455X (gfx1250) — compile-verified
//
#include <hip/hip_runtime.h>

typedef __attribute__((ext_vector_type(2))) float v2f;
typedef __attribute__((ext_vector_type(8))) float v8f;

#define BATCH   16384
#define NUMC    1000
#define FDIM    1024
#define MARGINF 5.0f

#define ROWS   32          // rows of A per workgroup (2 WMMA M-subtiles)
#define STRIDE 1028        // padded LDS row stride in floats (bank-conflict free)
#define NT     63          // 16-wide column tiles covering 1008 >= 1000 cols
#define WAVES  8

// ---------------------------------------------------------------- prep ------
__global__ __launch_bounds__(128)
void tcl_prep_kernel(const float* __restrict__ centers,
                     float* __restrict__ c2, int* __restrict__ iota,
                     float* __restrict__ out)
{
    __shared__ float s[128];
    const int c = blockIdx.x;
    const int t = threadIdx.x;
    const float* row = centers + (size_t)c * FDIM;
    float acc = 0.0f;
    for (int k = t; k < FDIM; k += 128) { float v = row[k]; acc += v * v; }
    s[t] = acc;
    __syncthreads();
    for (int o = 64; o > 0; o >>= 1) {
        if (t < o) s[t] += s[t + o];
        __syncthreads();
    }
    if (t == 0) {
        c2[c]   = s[0];
        iota[c] = c;
        if (c == 0) out[0] = 0.0f;   // re-zero accumulator every launch
    }
}

// ------------------------------------------------- fused GEMM + masked min --
// negout[r] = min over valid cols j (j < NUMC, j != labels[r]) of
//             0.5*c2[j] - dot(A[r], centers[j])
__global__ __launch_bounds__(256)
void tcl_gemm_min_kernel(const float* __restrict__ A, int nrows,
                         const int* __restrict__ labels,
                         const float* __restrict__ centers,
                         const float* __restrict__ c2,
                         float* __restrict__ negout)
{
    __shared__ float xs[ROWS * STRIDE];
    __shared__ int   lab[ROWS];
    __shared__ float red[WAVES][ROWS];

    const float INF = __builtin_inff();
    const int tid     = threadIdx.x;
    const int wave    = tid >> 5;
    const int lane    = tid & 31;
    const int rowbase = blockIdx.x * ROWS;

    // stage A tile (ROWS x FDIM) into padded LDS; zero-fill invalid rows
    for (int i = tid; i < ROWS * (FDIM / 4); i += 256) {
        const int r    = i / (FDIM / 4);
        const int cidx = (i % (FDIM / 4)) * 4;
        float4 v = make_float4(0.f, 0.f, 0.f, 0.f);
        const int gr = rowbase + r;
        if (gr < nrows) v = *(const float4*)(A + (size_t)gr * FDIM + cidx);
        *(float4*)(&xs[r * STRIDE + cidx]) = v;
    }
    if (tid < ROWS) {
        const int gr = rowbase + tid;
        lab[tid] = (gr < nrows) ? labels[gr] : -1;
    }
    __syncthreads();

    const int hi = lane >> 4;      // K-half select: lanes 0-15 -> K{0,1}, 16-31 -> K{2,3}
    const int ln = lane & 15;

    // hoist this lane's label values for the 16 C-fragment rows into registers
    // (row of C-fragment element e = e + 8*hi within each 16-row subtile)
    int labr0[8], labr1[8];
    #pragma unroll
    for (int e = 0; e < 8; ++e) {
        labr0[e] = lab[e + 8 * hi];
        labr1[e] = lab[16 + e + 8 * hi];
    }

    float rmin0[8], rmin1[8];
    #pragma unroll
    for (int e = 0; e < 8; ++e) { rmin0[e] = INF; rmin1[e] = INF; }

    const v2f* a0p = (const v2f*)(xs + (size_t)ln * STRIDE);
    const v2f* a1p = (const v2f*)(xs + (size_t)(16 + ln) * STRIDE);

    for (int ct = wave; ct < NT; ct += WAVES) {
        const int colbase = ct * 16;
        const int col     = colbase + ln;                 // this lane's N index
        const int ceff    = (col < NUMC) ? col : (NUMC - 1);
        const v2f* brow   = (const v2f*)(centers + (size_t)ceff * FDIM);

        v8f acc0 = {0.f,0.f,0.f,0.f,0.f,0.f,0.f,0.f};
        v8f acc1 = {0.f,0.f,0.f,0.f,0.f,0.f,0.f,0.f};

        #pragma unroll 8
        for (int t = 0; t < FDIM / 4; ++t) {
            const v2f b  = brow[2 * t + hi];   // B[4t+2h .. +1][col]
            const v2f a0 = a0p[2 * t + hi];    // A[row][4t+2h .. +1]
            const v2f a1 = a1p[2 * t + hi];
            acc0 = __builtin_amdgcn_wmma_f32_16x16x4_f32(
                       false, a0, false, b, (short)0, acc0, false, false);
            acc1 = __builtin_amdgcn_wmma_f32_16x16x4_f32(
                       false, a1, false, b, (short)0, acc1, false, false);
        }

        const float halfc2 = 0.5f * c2[ceff];
        const bool  colbad = (col >= NUMC);
        // branchless mask + running min (v_cmp / v_cndmask / v_min_num_f32)
        #pragma unroll
        for (int e = 0; e < 8; ++e) {
            const float s0 = halfc2 - acc0[e];
            const float s1 = halfc2 - acc1[e];
            const bool bad0 = colbad | (labr0[e] == col);
            const bool bad1 = colbad | (labr1[e] == col);
            const float m0 = bad0 ? INF : s0;
            const float m1 = bad1 ? INF : s1;
            rmin0[e] = fminf(rmin0[e], m0);
            rmin1[e] = fminf(rmin1[e], m1);
        }
    }

    // min across the 16 lanes of each half-wave (N dimension)
    #pragma unroll
    for (int e = 0; e < 8; ++e) {
        float v0 = rmin0[e], v1 = rmin1[e];
        for (int off = 1; off < 16; off <<= 1) {
            v0 = fminf(v0, __shfl_xor(v0, off, 16));
            v1 = fminf(v1, __shfl_xor(v1, off, 16));
        }
        if (ln == 0) {
            red[wave][e + 8 * hi]      = v0;
            red[wave][16 + e + 8 * hi] = v1;
        }
    }
    __syncthreads();

    if (wave == 0) {
        float v = INF;
        #pragma unroll
        for (int w = 0; w < WAVES; ++w) v = fminf(v, red[w][lane]);
        const int gr = rowbase + lane;
        if (gr < nrows) negout[gr] = v;
    }
}

// ------------------------------------------------------------- pos term -----
__global__ __launch_bounds__(256)
void tcl_pos_kernel(const float* __restrict__ x, const int* __restrict__ labels,
                    const float* __restrict__ centers, const float* __restrict__ c2,
                    float* __restrict__ pos_s)
{
    const int wave = threadIdx.x >> 5;
    const int lane = threadIdx.x & 31;
    const int i    = blockIdx.x * 8 + wave;
    const int lb   = labels[i];
    const float* xr = x + (size_t)i * FDIM;
    const float* cr = centers + (size_t)lb * FDIM;
    float acc = 0.0f;
    for (int k = lane; k < FDIM; k += 32) acc += xr[k] * cr[k];
    for (int off = 16; off > 0; off >>= 1) acc += __shfl_xor(acc, off, 32);
    if (lane == 0) pos_s[i] = 0.5f * c2[lb] - acc;
}

// ------------------------------------------------------------- final --------
__global__ __launch_bounds__(256)
void tcl_final_kernel(const int* __restrict__ labels, const float* __restrict__ c2,
                      const float* __restrict__ cenraw, const float* __restrict__ negraw,
                      const float* __restrict__ pos_s, float* __restrict__ out)
{
    __shared__ float s[256];
    const int i  = blockIdx.x * 256 + threadIdx.x;
    const int lb = labels[i];
    const float cen = 0.5f * c2[lb] + cenraw[lb];
    const float l = fmaxf(pos_s[i] - negraw[i] + MARGINF, 0.0f) +
                    fmaxf(MARGINF + 2.0f - cen, 0.0f);
    s[threadIdx.x] = l;
    __syncthreads();
    for (int o = 128; o > 0; o >>= 1) {
        if (threadIdx.x < o) s[threadIdx.x] += s[threadIdx.x + o];
        __syncthreads();
    }
    if (threadIdx.x == 0) atomicAdd(out, s[0] * (1.0f / BATCH));
}

// ------------------------------------------------------------- launch -------
extern "C" void kernel_launch(void* const* d_in, const int* in_sizes, int n_in,
                              void* d_out, int out_size, void* d_ws, size_t ws_size,
                              hipStream_t stream)
{
    const float* x       = (const float*)d_in[0];
    const int*   labels  = (const int*)d_in[1];
    const float* centers = (const float*)d_in[2];
    float* out = (float*)d_out;

    float* ws     = (float*)d_ws;
    float* c2     = ws;                       // 1024 floats
    int*   iota   = (int*)(ws + 1024);        // 1024 ints
    float* cenraw = ws + 2048;                // 1024 floats
    float* negraw = ws + 3072;                // 16384 floats
    float* pos_s  = ws + 3072 + BATCH;        // 16384 floats

    tcl_prep_kernel<<<NUMC, 128, 0, stream>>>(centers, c2, iota, out);
    tcl_gemm_min_kernel<<<(NUMC + ROWS - 1) / ROWS, 256, 0, stream>>>(
        centers, NUMC, iota, centers, c2, cenraw);
    tcl_gemm_min_kernel<<<BATCH / ROWS, 256, 0, stream>>>(
        x, BATCH, labels, centers, c2, negraw);
    tcl_pos_kernel<<<BATCH / 8, 256, 0, stream>>>(x, labels, centers, c2, pos_s);
    tcl_final_kernel<<<BATCH / 256, 256, 0, stream>>>(labels, c2, cenraw, negraw,
                                                      pos_s, out);
}